// MultiHeadTAP_46325517254988
// MI455X (gfx1250) — compile-verified
//
#include <hip/hip_runtime.h>
#include <math.h>

#define B_   64
#define T_   4096
#define C_   256
#define H_   4
#define HD_  64
#define NS_  8            // T-splits for the context pass
#define TS_  (T_ / NS_)   // 512
#define EPS_ 1e-5f

typedef float v2f __attribute__((ext_vector_type(2)));
typedef float v8f __attribute__((ext_vector_type(8)));

__device__ __forceinline__ void wait_asynccnt0() {
#if __has_builtin(__builtin_amdgcn_s_wait_asynccnt)
  __builtin_amdgcn_s_wait_asynccnt(0);
#else
  asm volatile("s_wait_asynccnt 0x0" ::: "memory");
#endif
}

// ---------------------------------------------------------------------------
// Kernel 1: scores[b,h,t] = w2[h] . tanh(x[b,t,:] @ W1[h] + b1[h])
// One wave handles a 16-row (b,t) tile; GEMM tile is M=16, N=256, K=256 via
// V_WMMA_F32_16X16X4_F32. A-fragments (16x256 of x) live in 64 v2f registers.
// The 256x16 B-tile of W1 is staged into LDS once per *workgroup* with the
// gfx1250 async global->LDS path (ASYNCcnt), then all 8 waves consume it with
// conflict-free ds_load_b32 (half0 banks lr/16+lr, half1 banks 32+lr/48+lr).
// ---------------------------------------------------------------------------
__global__ __launch_bounds__(256) void tap_scores_kernel(
    const float* __restrict__ x, const float* __restrict__ W1,
    const float* __restrict__ b1, const float* __restrict__ w2,
    float* __restrict__ scores)
{
  __shared__ float btile[C_ * 16];           // 16 KB: current B tile (K=256 x N=16)
  const int tid  = threadIdx.x;
  const int lane = tid & 31;
  const int wave = tid >> 5;
  const int lr   = lane & 15;
  const int half = lane >> 4;

  const int tile  = blockIdx.x * 8 + wave;   // 16-row tile id (16384 total)
  const int row0  = tile * 16;               // flat row = b*T + t
  const int b     = row0 >> 12;              // / T_
  const int tbase = row0 & (T_ - 1);

  // Stage this wave's A fragments (16 rows x 256 K) in registers: 64 x v2f.
  // A layout: lanes 0-15 -> (M=lane, K=k0+{0,1}); lanes 16-31 -> (M=lane-16, K=k0+{2,3}).
  const float* xrow = x + (size_t)(row0 + lr) * C_ + 2 * half;
  v2f a[64];
#pragma unroll
  for (int k = 0; k < 64; ++k)
    a[k] = *reinterpret_cast<const v2f*>(xrow + 4 * k);

  for (int h = 0; h < H_; ++h) {
    float sc[8] = {0.f, 0.f, 0.f, 0.f, 0.f, 0.f, 0.f, 0.f};
    for (int sub = 0; sub < 4; ++sub) {      // four 16-col tiles per head
      // ---- cooperative async stage of B tile: W1[h][:, sub*16 .. sub*16+16) ----
      // 1024 chunks of 16 B; each thread issues 4 async b128 copies.
      {
        const float* gbase = W1 + (size_t)h * C_ * HD_ + sub * 16;
#pragma unroll
        for (int it = 0; it < 4; ++it) {
          const int chunk = tid + it * 256;                   // 0..1023
          const float* gsrc = gbase + (size_t)(chunk >> 2) * HD_ + (chunk & 3) * 4;
          unsigned ldst = (unsigned)(size_t)(btile + chunk * 4);
          asm volatile("global_load_async_to_lds_b128 %0, %1, off"
                       :: "v"(ldst), "v"(gsrc) : "memory");
        }
        wait_asynccnt0();
      }
      __syncthreads();

      v8f acc = {};
#pragma unroll
      for (int k = 0; k < 64; ++k) {
        const int c0 = 4 * k + 2 * half;      // B: lane supplies (K=c0{,+1}, N=lr)
        v2f bf;
        bf.x = btile[c0 * 16 + lr];
        bf.y = btile[(c0 + 1) * 16 + lr];
        acc = __builtin_amdgcn_wmma_f32_16x16x4_f32(
            /*neg_a=*/false, a[k], /*neg_b=*/false, bf,
            /*c_mod=*/(short)0, acc, /*reuse_a=*/false, /*reuse_b=*/false);
      }

      const float b1v = b1[h * HD_ + sub * 16 + lr];
      const float w2v = w2[h * HD_ + sub * 16 + lr];
      // D layout: VGPR r / lanes 0-15 -> (M=r, N=lane); lanes 16-31 -> (M=r+8).
#pragma unroll
      for (int r = 0; r < 8; ++r) {
        float v = tanhf(acc[r] + b1v) * w2v;  // fuse activation + w2 dot
        v += __shfl_xor(v, 1, 32);            // reduce over the 16 columns
        v += __shfl_xor(v, 2, 32);            // (masks <16 stay inside half)
        v += __shfl_xor(v, 4, 32);
        v += __shfl_xor(v, 8, 32);
        sc[r] += v;
      }
      __syncthreads();                        // btile reuse fence
    }
    if (lr == 0) {  // lane 0 writes rows 0..7, lane 16 writes rows 8..15
      float* so = scores + (size_t)(b * H_ + h) * T_ + tbase + 8 * half;
#pragma unroll
      for (int r = 0; r < 8; ++r) so[r] = sc[r];
    }
  }
}

// ---------------------------------------------------------------------------
// Kernel 2: softmax over T for each (b,h) row. 256 threads, 16 vals/thread.
// ---------------------------------------------------------------------------
__global__ __launch_bounds__(256) void tap_softmax_kernel(
    const float* __restrict__ scores, float* __restrict__ weights)
{
  __shared__ float red[256];
  const int row = blockIdx.x;                 // b*H + h
  const int tid = threadIdx.x;
  const float* s = scores + (size_t)row * T_;

  float v[16];
  float mx = -3.402823466e38f;
#pragma unroll
  for (int i = 0; i < 16; ++i) { v[i] = s[tid + i * 256]; mx = fmaxf(mx, v[i]); }

  red[tid] = mx; __syncthreads();
  for (int off = 128; off > 0; off >>= 1) {
    if (tid < off) red[tid] = fmaxf(red[tid], red[tid + off]);
    __syncthreads();
  }
  mx = red[0]; __syncthreads();

  float sum = 0.f;
#pragma unroll
  for (int i = 0; i < 16; ++i) { v[i] = __expf(v[i] - mx); sum += v[i]; }
  red[tid] = sum; __syncthreads();
  for (int off = 128; off > 0; off >>= 1) {
    if (tid < off) red[tid] += red[tid + off];
    __syncthreads();
  }
  const float inv = 1.f / red[0];

  float* w = weights + (size_t)row * T_;
#pragma unroll
  for (int i = 0; i < 16; ++i) w[tid + i * 256] = v[i] * inv;
}

// ---------------------------------------------------------------------------
// Kernel 3: context partials. Block = (b, t-split); thread owns channel c and
// accumulates all 4 heads while streaming x once (weights staged in LDS).
// ---------------------------------------------------------------------------
__global__ __launch_bounds__(256) void tap_context_kernel(
    const float* __restrict__ x, const float* __restrict__ weights,
    float* __restrict__ partial)
{
  __shared__ float wsh[H_ * TS_];             // 4*512 floats = 8 KB
  const int b  = blockIdx.x / NS_;
  const int sp = blockIdx.x % NS_;
  const int tid = threadIdx.x;

  for (int idx = tid; idx < H_ * TS_; idx += 256) {
    const int h = idx / TS_, t = idx % TS_;
    wsh[idx] = weights[(size_t)(b * H_ + h) * T_ + sp * TS_ + t];
  }
  __syncthreads();

  const int c = tid;
  const float* xb = x + ((size_t)b * T_ + (size_t)sp * TS_) * C_ + c;
  float acc0 = 0.f, acc1 = 0.f, acc2 = 0.f, acc3 = 0.f;
  for (int t = 0; t < TS_; ++t) {
    if (t + 64 < TS_) __builtin_prefetch(xb + (size_t)(t + 64) * C_, 0, 0);
    const float xv = xb[(size_t)t * C_];
    acc0 = fmaf(wsh[0 * TS_ + t], xv, acc0);
    acc1 = fmaf(wsh[1 * TS_ + t], xv, acc1);
    acc2 = fmaf(wsh[2 * TS_ + t], xv, acc2);
    acc3 = fmaf(wsh[3 * TS_ + t], xv, acc3);
  }
  float* p = partial + ((size_t)(b * NS_ + sp)) * (H_ * C_) + c;
  p[0 * C_] = acc0; p[1 * C_] = acc1; p[2 * C_] = acc2; p[3 * C_] = acc3;
}

// ---------------------------------------------------------------------------
// Kernel 4: reduce partials -> multi (B,1024); out = multi @ Wo^T + bo; then
// LayerNorm over C. One block per b.
// ---------------------------------------------------------------------------
__global__ __launch_bounds__(256) void tap_output_kernel(
    const float* __restrict__ partial, const float* __restrict__ Wo,
    const float* __restrict__ bo, const float* __restrict__ gamma,
    const float* __restrict__ beta, float* __restrict__ out)
{
  __shared__ float multi_s[H_ * C_];          // 1024 floats
  __shared__ float red[256];
  const int b = blockIdx.x;
  const int tid = threadIdx.x;

  for (int j = tid; j < H_ * C_; j += 256) {
    float a = 0.f;
    for (int sp = 0; sp < NS_; ++sp)
      a += partial[((size_t)(b * NS_ + sp)) * (H_ * C_) + j];
    multi_s[j] = a;
  }
  __syncthreads();

  const int c = tid;
  float acc = bo[c];
  const float* wrow = Wo + (size_t)c * (H_ * C_);
#pragma unroll 8
  for (int j = 0; j < H_ * C_; ++j) acc = fmaf(multi_s[j], wrow[j], acc);

  // LayerNorm over the 256 channels of this row
  red[tid] = acc; __syncthreads();
  for (int off = 128; off > 0; off >>= 1) {
    if (tid < off) red[tid] += red[tid + off];
    __syncthreads();
  }
  const float mu = red[0] * (1.f / C_); __syncthreads();

  const float d = acc - mu;
  red[tid] = d * d; __syncthreads();
  for (int off = 128; off > 0; off >>= 1) {
    if (tid < off) red[tid] += red[tid + off];
    __syncthreads();
  }
  const float var = red[0] * (1.f / C_);

  out[(size_t)b * C_ + c] = d * rsqrtf(var + EPS_) * gamma[c] + beta[c];
}

// ---------------------------------------------------------------------------
extern "C" void kernel_launch(void* const* d_in, const int* in_sizes, int n_in,
                              void* d_out, int out_size, void* d_ws, size_t ws_size,
                              hipStream_t stream) {
  (void)in_sizes; (void)n_in; (void)out_size; (void)ws_size;

  const float* x     = (const float*)d_in[0];
  const float* W1    = (const float*)d_in[1];
  const float* b1    = (const float*)d_in[2];
  const float* w2    = (const float*)d_in[3];
  const float* Wo    = (const float*)d_in[4];
  const float* bo    = (const float*)d_in[5];
  const float* gamma = (const float*)d_in[6];
  const float* beta  = (const float*)d_in[7];

  float* out     = (float*)d_out;                 // (B, C)
  float* weights = out + (size_t)B_ * C_;         // (B, H, T) second output

  float* scores  = (float*)d_ws;                  // B*H*T   = 1,048,576 floats
  float* partial = scores + (size_t)B_ * H_ * T_; // B*NS*H*C =  524,288 floats

  // K1: 16384 wave-tiles / 8 waves per block
  tap_scores_kernel<<<(B_ * T_ / 16) / 8, 256, 0, stream>>>(x, W1, b1, w2, scores);
  // K2: one block per (b,h)
  tap_softmax_kernel<<<B_ * H_, 256, 0, stream>>>(scores, weights);
  // K3: one block per (b, t-split)
  tap_context_kernel<<<B_ * NS_, 256, 0, stream>>>(x, weights, partial);
  // K4: one block per b
  tap_output_kernel<<<B_, 256, 0, stream>>>(partial, Wo, bo, gamma, beta, out);
}